// adder2d_24472723653052
// MI455X (gfx1250) — compile-verified
//
#include <hip/hip_runtime.h>
#include <cstdint>
#include <cstddef>

// AdderNet 2D: out[n,f,oh,ow] = -sum_{c,kh,kw} |W[f,c,kh,kw] - xpad[n,c,oh+kh-1,ow+kw-1]|
// N=4, C=64, H=W=28, F=64, K=3, stride 1, pad 1.

#define N_   4
#define C_   64
#define HW_  28
#define F_   64
#define D_   576          // C_ * 9
#define FT   16           // filters per block
#define OT   8            // output tile: 8 wide x 8 tall
#define PT   10           // patch width/height (OT + 2)
#define XSTR 12           // padded LDS row stride for x patch (16B-aligned rows)
#define CB   32           // channels staged per chunk (2 chunks)
#define WSTR 577          // padded LDS row stride for W (odd -> bank-conflict-free)

#ifndef __has_builtin
#define __has_builtin(x) 0
#endif

#if defined(__gfx1250__) && __has_builtin(__builtin_amdgcn_tensor_load_to_lds)
#define ADDER_TDM 1
#if defined(__has_include)
#if __has_include(<hip/amd_detail/amd_gfx1250_TDM.h>)
#define ADDER_TDM6 1   // amdgpu-toolchain (clang-23): 6-arg builtin
#endif
#endif
#endif

typedef unsigned int u32x4_t __attribute__((ext_vector_type(4)));
typedef int          i32x4_t __attribute__((ext_vector_type(4)));
typedef int          i32x8_t __attribute__((ext_vector_type(8)));

__global__ __launch_bounds__(128) void adder2d_24472723653052_kernel(
    const float* __restrict__ X, const float* __restrict__ Wt,
    float* __restrict__ Out) {
  __shared__ float Wl[FT * WSTR];                       // 36,928 B (TDM-filled, padded)
  __shared__ __align__(16) float Xl[CB * PT * XSTR];    // 15,360 B per channel chunk

  const int tid = threadIdx.x;           // 128 threads = 16 filters x 8 rows
  const int st  = blockIdx.x;            // 0..15 spatial tile
  const int tx  = st & 3, ty = st >> 2;
  const int fb  = blockIdx.y * FT;       // filter base
  const int n   = blockIdx.z;
  const int ox0 = tx * OT, oy0 = ty * OT;

#if ADDER_TDM
  if (tid < 32) {
    // Tensor Data Mover: W[fb .. fb+15, 0:576] -> LDS with row stride 577
    // via the iterate feature (16 iterations of a 576-element row).
    uint64_t gaddr = (uint64_t)(uintptr_t)(Wt + (size_t)fb * D_);
    uint32_t laddr = (uint32_t)(uintptr_t)(&Wl[0]); // flat addr low 32b == LDS offset

    u32x4_t g0;
    g0[0] = 1u;                                        // count=1, user descriptor
    g0[1] = laddr;                                     // lds_addr (bytes)
    g0[2] = (uint32_t)gaddr;                           // global_addr[31:0]
    g0[3] = ((uint32_t)(gaddr >> 32) & 0x01FFFFFFu)    // global_addr[56:32]
            | (2u << 30);                              // type = 2 ("image")

    i32x8_t g1;
    g1[0] = (int)((2u << 16) | (1u << 19));            // data_size=4B, iterate_enable
    g1[1] = (int)((uint32_t)(D_ * FT) << 16);          // tensor_dim0[15:0] -> bits[63:48]
    g1[2] = (int)(((uint32_t)(D_ * FT) >> 16)          // tensor_dim0[31:16]
                  | (64u << 16));                      // tensor_dim1 = 64
    g1[3] = (int)((uint32_t)D_ << 16);                 // tile_dim0 = 576
    g1[4] = 1;                                         // tile_dim1 = 1, tile_dim2 = 0
    g1[5] = D_;                                        // tensor_dim0_stride[31:0]
    g1[6] = (int)((uint32_t)D_ << 16);                 // dim0_stride hi=0 | dim1_stride=576
    g1[7] = 0;

    i32x4_t g2;
    g2[0] = 1;                                         // tensor_dim2 (unused)
    g2[1] = WSTR;                                      // lds_addr_increment / iter (elems)
    g2[2] = D_;                                        // global_addr_increment / iter
    g2[3] = (int)((uint32_t)(FT - 1) << 16);           // iterate_count = 15 -> 16 iters
    i32x4_t g3; g3[0] = 0; g3[1] = 0; g3[2] = 0; g3[3] = 0;

#if ADDER_TDM6
    i32x8_t g4; g4[0]=0; g4[1]=0; g4[2]=0; g4[3]=0; g4[4]=0; g4[5]=0; g4[6]=0; g4[7]=0;
    __builtin_amdgcn_tensor_load_to_lds(g0, g1, g2, g3, g4, 0);
#else
    __builtin_amdgcn_tensor_load_to_lds(g0, g1, g2, g3, 0);
#endif
    __builtin_amdgcn_s_wait_tensorcnt(0);
  }
#else
  for (int e = tid; e < FT * D_; e += 128) {
    int f = e / D_, d = e - f * D_;
    Wl[f * WSTR + d] = Wt[(size_t)(fb + f) * D_ + d];
  }
#endif

  const int fl = tid & (FT - 1);         // filter lane 0..15
  const int py = tid >> 4;               // output row within tile 0..7

  float a[OT];
#pragma unroll
  for (int i = 0; i < OT; ++i) a[i] = 0.0f;

  const float* xn = X + (size_t)n * C_ * HW_ * HW_;

  for (int cb = 0; cb < C_; cb += CB) {  // two channel chunks
    __syncthreads();                     // previous chunk fully consumed / TDM ordered
    // Stage zero-padded x patch for channels [cb, cb+CB): rows oy0-1..oy0+8,
    // cols ox0-1..ox0+8, padded to a 12-float row stride (cols 10..11 unused).
    for (int e = tid; e < CB * PT * XSTR; e += 128) {
      int c  = e / (PT * XSTR);
      int r  = e - c * (PT * XSTR);
      int ry = r / XSTR, rx = r - ry * XSTR;
      int iy = oy0 - 1 + ry, ix = ox0 - 1 + rx;
      float v = 0.0f;
      if (rx < PT && (unsigned)iy < (unsigned)HW_ && (unsigned)ix < (unsigned)HW_)
        v = xn[((cb + c) * HW_ + iy) * HW_ + ix];
      Xl[e] = v;
    }
    __syncthreads();

    for (int c = 0; c < CB; ++c) {
      const float* wc = &Wl[fl * WSTR + (cb + c) * 9];
      const float* xb = &Xl[c * (PT * XSTR)];
#pragma unroll
      for (int kh = 0; kh < 3; ++kh) {
        const float* xr = xb + (py + kh) * XSTR;       // 16B-aligned row
        float4 v0 = *(const float4*)(xr);              // ds_load_b128
        float4 v1 = *(const float4*)(xr + 4);          // ds_load_b128
        float2 v2 = *(const float2*)(xr + 8);          // ds_load_b64
        float xs[10] = {v0.x, v0.y, v0.z, v0.w, v1.x, v1.y, v1.z, v1.w, v2.x, v2.y};
        float w0 = wc[kh * 3 + 0];
        float w1 = wc[kh * 3 + 1];
        float w2 = wc[kh * 3 + 2];
#pragma unroll
        for (int i = 0; i < OT; ++i)
          a[i] += __builtin_fabsf(w0 - xs[i]) +
                  __builtin_fabsf(w1 - xs[i + 1]) +
                  __builtin_fabsf(w2 - xs[i + 2]);
      }
    }
  }

  const int oh = oy0 + py;
  if (oh < HW_) {
    float* orow = Out + (((size_t)n * F_ + (fb + fl)) * HW_ + oh) * HW_ + ox0;
#pragma unroll
    for (int i = 0; i < OT; ++i)
      if (ox0 + i < HW_) orow[i] = -a[i];
  }
}

extern "C" void kernel_launch(void* const* d_in, const int* in_sizes, int n_in,
                              void* d_out, int out_size, void* d_ws, size_t ws_size,
                              hipStream_t stream) {
  const float* x  = (const float*)d_in[0];   // (4, 64, 28, 28) f32
  const float* w  = (const float*)d_in[1];   // (64, 64, 3, 3)  f32
  float* out      = (float*)d_out;           // (4, 64, 28, 28) f32
  (void)in_sizes; (void)n_in; (void)out_size; (void)d_ws; (void)ws_size;

  dim3 grid(16, 4, 4);                       // spatial tiles, f-tiles, batch
  adder2d_24472723653052_kernel<<<grid, dim3(128), 0, stream>>>(x, w, out);
}